// GaussianRasterizer_30124900614753
// MI455X (gfx1250) — compile-verified
//
#include <hip/hip_runtime.h>
#include <hip/hip_bf16.h>
#include <stdint.h>

// ---------------- problem constants (match reference) ----------------
#define WIMG   256
#define HIMG   256
#define HWPIX  (WIMG * HIMG)
#define N_G    8192
#define CHUNK  128
#define NCHUNK (N_G / CHUNK)
#define REC    16            // floats per gaussian record (64 B)
#define TANX   0.5773502691896258f
#define TANY   0.5773502691896258f
#define SCALE_MOD 1.0f

typedef float    v2f __attribute__((ext_vector_type(2)));
typedef float    v8f __attribute__((ext_vector_type(8)));
typedef unsigned int v4u __attribute__((ext_vector_type(4)));
typedef int      v4i __attribute__((ext_vector_type(4)));
typedef int      v8i __attribute__((ext_vector_type(8)));

// =====================================================================
// Kernel 1: per-gaussian preprocessing (projection, conic, radius,
// quadratic-form features G0..G5 for the WMMA power computation).
// =====================================================================
__global__ __launch_bounds__(256) void gs_preprocess(
    const float* __restrict__ m3d, const float* __restrict__ opac,
    const float* __restrict__ imps, const float* __restrict__ cols,
    const float* __restrict__ scls, const float* __restrict__ rots,
    const float* __restrict__ vm,   const float* __restrict__ pm,
    float* __restrict__ keys, float* __restrict__ recs,
    int* __restrict__ radii_out)
{
    int i = blockIdx.x * 256 + threadIdx.x;
    if (i >= N_G) return;

    float x = m3d[i*3+0], y = m3d[i*3+1], z = m3d[i*3+2];
    float pv[3], hom[4];
    #pragma unroll
    for (int j = 0; j < 3; ++j) pv[j]  = x*vm[j] + y*vm[4+j] + z*vm[8+j] + vm[12+j];
    #pragma unroll
    for (int j = 0; j < 4; ++j) hom[j] = x*pm[j] + y*pm[4+j] + z*pm[8+j] + pm[12+j];

    float depth = pv[2];
    float pwv = 1.0f / (hom[3] + 1e-7f);
    float px = ((hom[0]*pwv + 1.0f) * WIMG - 1.0f) * 0.5f;
    float py = ((hom[1]*pwv + 1.0f) * HIMG - 1.0f) * 0.5f;

    // quaternion -> rotation
    float qr = rots[i*4+0], qx = rots[i*4+1], qy = rots[i*4+2], qz = rots[i*4+3];
    float qn = rsqrtf(qr*qr + qx*qx + qy*qy + qz*qz);
    qr *= qn; qx *= qn; qy *= qn; qz *= qn;
    float R00 = 1.f-2.f*(qy*qy+qz*qz), R01 = 2.f*(qx*qy-qr*qz), R02 = 2.f*(qx*qz+qr*qy);
    float R10 = 2.f*(qx*qy+qr*qz), R11 = 1.f-2.f*(qx*qx+qz*qz), R12 = 2.f*(qy*qz-qr*qx);
    float R20 = 2.f*(qx*qz-qr*qy), R21 = 2.f*(qy*qz+qr*qx), R22 = 1.f-2.f*(qx*qx+qy*qy);

    float sx = scls[i*3+0]*SCALE_MOD, sy = scls[i*3+1]*SCALE_MOD, sz = scls[i*3+2]*SCALE_MOD;
    float s2x = sx*sx, s2y = sy*sy, s2z = sz*sz;
    float S00 = R00*R00*s2x + R01*R01*s2y + R02*R02*s2z;
    float S01 = R00*R10*s2x + R01*R11*s2y + R02*R12*s2z;
    float S02 = R00*R20*s2x + R01*R21*s2y + R02*R22*s2z;
    float S11 = R10*R10*s2x + R11*R11*s2y + R12*R12*s2z;
    float S12 = R10*R20*s2x + R11*R21*s2y + R12*R22*s2z;
    float S22 = R20*R20*s2x + R21*R21*s2y + R22*R22*s2z;

    const float fx = WIMG / (2.0f*TANX), fy = HIMG / (2.0f*TANY);
    const float limx = 1.3f*TANX, limy = 1.3f*TANY;
    float tz  = depth, tzi = 1.0f / tz;
    float txc = fminf(fmaxf(pv[0]*tzi, -limx), limx) * tz;
    float tyc = fminf(fmaxf(pv[1]*tzi, -limy), limy) * tz;
    float j00 = fx*tzi, j02 = -fx*txc*tzi*tzi;
    float j11 = fy*tzi, j12 = -fy*tyc*tzi*tzi;

    float M0[3], M1[3];
    #pragma unroll
    for (int c = 0; c < 3; ++c) {
        M0[c] = j00*vm[c*4+0] + j02*vm[c*4+2];   // J @ viewmatrix[:3,:3].T
        M1[c] = j11*vm[c*4+1] + j12*vm[c*4+2];
    }
    float t0x = S00*M0[0] + S01*M0[1] + S02*M0[2];
    float t0y = S01*M0[0] + S11*M0[1] + S12*M0[2];
    float t0z = S02*M0[0] + S12*M0[1] + S22*M0[2];
    float cov00 = M0[0]*t0x + M0[1]*t0y + M0[2]*t0z;
    float cov01 = M1[0]*t0x + M1[1]*t0y + M1[2]*t0z;
    float t1x = S00*M1[0] + S01*M1[1] + S02*M1[2];
    float t1y = S01*M1[0] + S11*M1[1] + S12*M1[2];
    float t1z = S02*M1[0] + S12*M1[1] + S22*M1[2];
    float cov11 = M1[0]*t1x + M1[1]*t1y + M1[2]*t1z;

    float a = cov00 + 0.3f, cc = cov11 + 0.3f, b = cov01;
    float det = a*cc - b*b;
    float det_safe = (det != 0.0f) ? det : 1.0f;
    float co0 = cc / det_safe, co1 = -b / det_safe, co2 = a / det_safe;
    float mid = 0.5f * (a + cc);
    float lam = mid + sqrtf(fmaxf(0.1f, mid*mid - det));
    float radius = ceilf(3.0f * sqrtf(fmaxf(lam, 0.0f)));
    bool valid = (depth > 0.2f) && (det > 0.0f) && (radius > 0.0f);

    radii_out[i] = valid ? (int)radius : 0;
    keys[i] = valid ? depth : __builtin_inff();

    float* r = recs + (size_t)i * REC;
    r[0] = px; r[1] = py; r[2] = opac[i]; r[3] = radius;
    r[4] = cols[i*3+0]; r[5] = cols[i*3+1]; r[6] = cols[i*3+2]; r[7] = imps[i];
    // power(g; X,Y) = G . [1, X, Y, X^2, Y^2, X*Y]
    r[8]  = -0.5f*(co0*px*px + co2*py*py) - co1*px*py;
    r[9]  = co0*px + co1*py;
    r[10] = co2*py + co1*px;
    r[11] = -0.5f*co0;
    r[12] = -0.5f*co2;
    r[13] = -co1;
    r[14] = valid ? 1.0f : 0.0f;
    r[15] = 0.0f;
}

// =====================================================================
// Kernel 2: single-workgroup LDS bitonic argsort by depth (8192 keys).
// 64 KB LDS, well within the 320 KB WGP budget.
// =====================================================================
__global__ __launch_bounds__(1024) void gs_sort(const float* __restrict__ keys,
                                                int* __restrict__ order)
{
    __shared__ float sk[N_G];
    __shared__ int   sv[N_G];
    const int tid = threadIdx.x;
    for (int i = tid; i < N_G; i += 1024) { sk[i] = keys[i]; sv[i] = i; }
    __syncthreads();
    for (int len = 2; len <= N_G; len <<= 1) {
        for (int j = len >> 1; j > 0; j >>= 1) {
            for (int t = tid; t < N_G/2; t += 1024) {
                int lo = t & (j - 1);
                int i  = ((t - lo) << 1) | lo;
                int p  = i | j;
                bool up = ((i & len) == 0);
                float ki = sk[i], kp = sk[p];
                bool sw = up ? (ki > kp) : (ki < kp);
                if (sw) {
                    int vi = sv[i];
                    sk[i] = kp; sk[p] = ki;
                    sv[i] = sv[p]; sv[p] = vi;
                }
            }
            __syncthreads();
        }
    }
    for (int i = tid; i < N_G; i += 1024) order[i] = sv[i];
}

// =====================================================================
// Kernel 3: gather records into depth order (chunk-contiguous for TDM).
// =====================================================================
__global__ __launch_bounds__(256) void gs_gather(const float* __restrict__ recs,
                                                 const int* __restrict__ order,
                                                 float* __restrict__ srec)
{
    int j = blockIdx.x * 256 + threadIdx.x;
    if (j >= N_G) return;
    int g = order[j];
    const float4* s = (const float4*)(recs + (size_t)g * REC);
    float4*       d = (float4*)(srec + (size_t)j * REC);
    d[0] = s[0]; d[1] = s[1]; d[2] = s[2]; d[3] = s[3];
}

// =====================================================================
// TDM: async 1-D tile load of one 8 KB chunk (128 records) into LDS.
// D# group0: count=1, lds_addr, 57-bit global addr, type=2.
// D# group1: data_size=4B, tensor_dim0=tile_dim0=2048, dim1=1, stride=2048.
// 6-arg builtin form: (g0, g1, g2, g3, g_extra, cpol); groups 2/3 zero
// (<=2-D tensor).
// =====================================================================
__device__ __forceinline__ void tdm_load_chunk(const float* gsrc, unsigned lds_off)
{
    uint64_t ga = (uint64_t)(uintptr_t)gsrc;
    v4u g0;
    g0.x = 1u;                                              // count = 1 valid descriptor
    g0.y = lds_off;                                         // LDS byte address
    g0.z = (unsigned)(ga & 0xFFFFFFFFu);                    // global addr [31:0]
    g0.w = (unsigned)((ga >> 32) & 0x01FFFFFFu) | (2u << 30); // addr[56:32] | type=2
    v8i g1;
    g1[0] = (int)(2u << 16);       // workgroup_mask=0, data_size=2 (4 bytes)
    g1[1] = (int)(2048u << 16);    // tensor_dim0[15:0]=2048 in bits[63:48]
    g1[2] = (int)(1u << 16);       // tensor_dim0 hi=0 | tensor_dim1=1 (low bits)
    g1[3] = (int)(2048u << 16);    // tensor_dim1 hi=0 | tile_dim0=2048
    g1[4] = 1;                     // tile_dim1=1, tile_dim2=0
    g1[5] = 2048;                  // tensor_dim0_stride low32
    g1[6] = 0;
    g1[7] = 0;
    v4i zz4 = {0, 0, 0, 0};
    v8i zz8 = {0, 0, 0, 0, 0, 0, 0, 0};
    __builtin_amdgcn_tensor_load_to_lds(g0, g1, zz4, zz4, zz8, 0);
}

// =====================================================================
// Kernel 4: render. 256 threads = 8 waves; each wave owns 16 pixels.
// Per 16-gaussian tile: power tile via 2x V_WMMA_F32_16X16X4_F32
// (co-executes with the VALU blend), then parallel half-wave prefix
// blend combined exactly with shfl_xor(16). Chunks double-buffered in
// LDS via the Tensor Data Mover (TENSORcnt + barriers).
// =====================================================================
__global__ __launch_bounds__(256) void gs_render(const float* __restrict__ srec,
                                                 const float* __restrict__ bg,
                                                 float* __restrict__ outC,
                                                 float* __restrict__ outImp)
{
    __shared__ float lbuf[2][CHUNK * REC];   // 2 x 8 KB

    const int tid  = threadIdx.x;
    const int lane = tid & 31;
    const int wave = tid >> 5;
    const int sub  = lane & 15;              // N column / M row within tile
    const bool hi  = lane >= 16;

    const int pix = blockIdx.x * 128 + wave * 16 + sub;   // 512 blocks x 128 pixels
    const float X = (float)(pix & (WIMG - 1));
    const float Y = (float)(pix >> 8);

    // B operand (pixel features), K rows {1, X, Y, X^2} and {Y^2, XY, 0, 0}.
    // f32 4x16 layout: V0 = K(hi?2:0), V1 = K(hi?3:1) striped over lane halves.
    v2f b0, b1;
    b0.x = hi ? Y       : 1.0f;
    b0.y = hi ? X * X   : X;
    b1.x = hi ? 0.0f    : Y * Y;
    b1.y = hi ? 0.0f    : X * Y;

    float T = 1.0f, aR = 0.f, aG = 0.f, aB = 0.f, aI = 0.f;

    unsigned lds0 = (unsigned)(uintptr_t)&lbuf[0][0];
    unsigned lds1 = (unsigned)(uintptr_t)&lbuf[1][0];

    if (wave == 0) tdm_load_chunk(srec, lds0);

    for (int c = 0; c < NCHUNK; ++c) {
        if (wave == 0) {
            if (c + 1 < NCHUNK) {
                tdm_load_chunk(srec + (size_t)(c + 1) * CHUNK * REC,
                               ((c + 1) & 1) ? lds1 : lds0);
                __builtin_amdgcn_s_wait_tensorcnt(1);   // chunk c landed, c+1 in flight
            } else {
                __builtin_amdgcn_s_wait_tensorcnt(0);
            }
        }
        __syncthreads();

        const float* rec = &lbuf[c & 1][0];
        if (__any(T > 1e-4f)) {                         // wave-uniform early out
            #pragma unroll 2
            for (int t = 0; t < 8; ++t) {
                // A operand: gaussian features, 16x4 f32 layout
                // lane m (low): A[m][0..1]; lane m+16: A[m][2..3]
                const float* ab = rec + (size_t)(t * 16 + sub) * REC;
                v2f a0, a1;
                if (!hi) { a0.x = ab[8];  a0.y = ab[9];  a1.x = ab[12]; a1.y = ab[13]; }
                else     { a0.x = ab[10]; a0.y = ab[11]; a1.x = 0.0f;   a1.y = 0.0f;   }

                v8f pw = {0.f,0.f,0.f,0.f,0.f,0.f,0.f,0.f};
                pw = __builtin_amdgcn_wmma_f32_16x16x4_f32(false, a0, false, b0,
                                                           (short)0, pw, false, false);
                pw = __builtin_amdgcn_wmma_f32_16x16x4_f32(false, a1, false, b1,
                                                           (short)0, pw, false, false);

                // Sequential blend of this lane-half's 8 gaussians (local prefix)
                float Sr = 0.f, Sg = 0.f, Sb = 0.f, Si = 0.f, P = 1.0f;
                const float* gb = rec + (size_t)(t * 16 + (hi ? 8 : 0)) * REC;
                #pragma unroll
                for (int v = 0; v < 8; ++v) {
                    const float* g = gb + v * REC;
                    float4 p0 = *(const float4*)g;        // mx, my, opc, rad
                    float4 p1 = *(const float4*)(g + 4);  // r, g, b, imp
                    float vfl = g[14];
                    float dx = p0.x - X, dy = p0.y - Y;
                    float pwr = pw[v];
                    float al = fminf(0.99f, p0.z * __expf(pwr));
                    bool ok = (vfl != 0.0f) & (pwr <= 0.0f) & (al >= (1.0f/255.0f))
                            & (fabsf(dx) <= p0.w) & (fabsf(dy) <= p0.w);
                    al = ok ? al : 0.0f;
                    float Trun = T * P;
                    al = (Trun * (1.0f - al) < 1e-4f) ? 0.0f : al;
                    float w = al * P;
                    Sr += w * p1.x; Sg += w * p1.y; Sb += w * p1.z; Si += w * p1.w;
                    P *= (1.0f - al);
                }
                // Combine halves: full = S_lo + P_lo * S_hi ; T *= P_lo * P_hi
                float oSr = __shfl_xor(Sr, 16), oSg = __shfl_xor(Sg, 16),
                      oSb = __shfl_xor(Sb, 16), oSi = __shfl_xor(Si, 16),
                      oP  = __shfl_xor(P, 16);
                float fr = hi ? (oSr + oP * Sr) : (Sr + P * oSr);
                float fg = hi ? (oSg + oP * Sg) : (Sg + P * oSg);
                float fb = hi ? (oSb + oP * Sb) : (Sb + P * oSb);
                float fi = hi ? (oSi + oP * Si) : (Si + P * oSi);
                aR += T * fr; aG += T * fg; aB += T * fb; aI += T * fi;
                T  *= P * oP;
            }
        }
        __syncthreads();
    }

    if (!hi) {   // both halves hold identical results; low half stores
        outC[0 * HWPIX + pix] = aR + T * bg[0];
        outC[1 * HWPIX + pix] = aG + T * bg[1];
        outC[2 * HWPIX + pix] = aB + T * bg[2];
        outImp[pix] = aI;
    }
}

// =====================================================================
// Host launcher
// =====================================================================
extern "C" void kernel_launch(void* const* d_in, const int* in_sizes, int n_in,
                              void* d_out, int out_size, void* d_ws, size_t ws_size,
                              hipStream_t stream)
{
    (void)in_sizes; (void)n_in; (void)out_size; (void)ws_size;
    const float* means3D = (const float*)d_in[0];
    // d_in[1] = means2D (unused by reference math)
    const float* opac    = (const float*)d_in[2];
    const float* imps    = (const float*)d_in[3];
    const float* cols    = (const float*)d_in[4];
    const float* scls    = (const float*)d_in[5];
    const float* rots    = (const float*)d_in[6];
    const float* bg      = (const float*)d_in[7];
    const float* vm      = (const float*)d_in[8];
    const float* pm      = (const float*)d_in[9];

    float* ws    = (float*)d_ws;
    float* keys  = ws;                                  // 8192
    float* recs  = ws + N_G;                            // 8192*16 (unsorted)
    float* srec  = ws + N_G + (size_t)N_G * REC;        // 8192*16 (sorted)
    int*   order = (int*)(ws + N_G + 2 * (size_t)N_G * REC); // 8192

    float* outC   = (float*)d_out;                      // (3, H, W)
    int*   outRad = (int*)d_out + 3 * HWPIX;            // (N,) int32
    float* outImp = (float*)d_out + 3 * HWPIX + N_G;    // (H, W)

    gs_preprocess<<<N_G / 256, 256, 0, stream>>>(means3D, opac, imps, cols, scls,
                                                 rots, vm, pm, keys, recs, outRad);
    gs_sort<<<1, 1024, 0, stream>>>(keys, order);
    gs_gather<<<N_G / 256, 256, 0, stream>>>(recs, order, srec);
    gs_render<<<HWPIX / 128, 256, 0, stream>>>(srec, bg, outC, outImp);
}